// MultiTaskLoss_89000312307783
// MI455X (gfx1250) — compile-verified
//
#include <hip/hip_runtime.h>

typedef float v2f __attribute__((ext_vector_type(2)));
typedef float v4f __attribute__((ext_vector_type(4)));
typedef float v8f __attribute__((ext_vector_type(8)));

#define MTL_DT     0.01f
#define MTL_MINVAR 0.01f
#define MTL_EPS    1e-6f
#define B_DIM 256
#define T_DIM 16384

// One block per batch row b. 256 threads = 8 wave32s.
// Each iteration: wave w processes the 256-timestep tile (it*8 + w).
// Exclusive scan of a 16x16 tile M is computed as D = M * U' + C where
// U' is strict-upper-triangular ones (K=16 via 4 chained f32 16x16x4 WMMAs)
// and C broadcasts vel[0] + carry + wave offset + row offsets.
// All bulk traffic is single-use streaming -> non-temporal loads.
__global__ __launch_bounds__(256) void mtl_main_kernel(
    const float* __restrict__ vel_pred,
    const float* __restrict__ cov_pred,
    const float* __restrict__ vel_gt,
    const float* __restrict__ pos_gt,
    float* __restrict__ ws)
{
    const int b       = blockIdx.x;
    const int tid     = threadIdx.x;
    const int wave    = tid >> 5;
    const int lane    = tid & 31;
    const int lane16  = lane & 15;
    const int khalf   = (lane < 16) ? 0 : 2;   // A/B layout: K offset per lane half
    const int rowbase = (lane < 16) ? 0 : 8;   // C/D layout: M row base per lane half

    const size_t base = (size_t)b * T_DIM;     // timestep index offset for this row
    const v4f* __restrict__ vel4 = (const v4f*)vel_pred;
    const v4f* __restrict__ gt4  = (const v4f*)vel_gt;
    const v4f* __restrict__ cov4 = (const v4f*)cov_pred;
    const v2f* __restrict__ pg2  = (const v2f*)pos_gt;

    const v2f v0 = ((const v2f*)vel_pred)[base];         // vel_pred[b, 0, :]

    float accp = 0.f, acce = 0.f, accml = 0.f, accreg = 0.f;
    float carry_x = 0.f, carry_y = 0.f;                  // running total of all prior tiles

    __shared__ float wsx[2][8];   // per-wave tile totals (double buffered)
    __shared__ float wsy[2][8];

    for (int it = 0; it < 8; ++it) {
        const int tb = (it * 8 + wave) * 256;            // tile base timestep

        // ---- load tile in f32 A-matrix layout; fuse elementwise terms ----
        v2f ax[4], ay[4];
        float rpx = 0.f, rpy = 0.f;                      // half-row partial sums
        #pragma unroll
        for (int c = 0; c < 4; ++c) {
            const int t = tb + 16 * lane16 + 4 * c + khalf;   // this lane's (row, k) element
            const size_t q = (base + (size_t)t) >> 1;         // v4f index ({x,y} x 2 steps)
            v4f fv = __builtin_nontemporal_load(vel4 + q);
            v4f fg = __builtin_nontemporal_load(gt4 + q);
            v4f fc = __builtin_nontemporal_load(cov4 + q);
            ax[c].x = fv.x; ax[c].y = fv.z;                   // x-component A regs (K=k, K=k+1)
            ay[c].x = fv.y; ay[c].y = fv.w;                   // y-component A regs
            rpx += fv.x + fv.z;
            rpy += fv.y + fv.w;
            {   // timestep t
                float dxv = fg.x - fv.x, dyv = fg.y - fv.y;
                acce += dxv * dxv + dyv * dyv;
                float sx = fmaxf(__expf(fc.x), MTL_MINVAR);
                float sy = fmaxf(__expf(fc.y), MTL_MINVAR);
                float ix = 1.f / sx, iy = 1.f / sy;
                accml += ix * dxv * dxv + iy * dyv * dyv + __logf(sx * sy + MTL_EPS);
                accreg += ix + iy;
            }
            {   // timestep t+1
                float dxv = fg.z - fv.z, dyv = fg.w - fv.w;
                acce += dxv * dxv + dyv * dyv;
                float sx = fmaxf(__expf(fc.z), MTL_MINVAR);
                float sy = fmaxf(__expf(fc.w), MTL_MINVAR);
                float ix = 1.f / sx, iy = 1.f / sy;
                accml += ix * dxv * dxv + iy * dyv * dyv + __logf(sx * sy + MTL_EPS);
                accreg += ix + iy;
            }
        }

        // ---- full row sums (lane L and L+16 hold the two halves of row L%16) ----
        float rx = rpx + __shfl_xor(rpx, 16, 32);
        float ry = rpy + __shfl_xor(rpy, 16, 32);

        // wave (tile) totals: reduce the 16 row sums (duplicated per half)
        float wx = rx, wy = ry;
        #pragma unroll
        for (int m = 1; m < 16; m <<= 1) {
            wx += __shfl_xor(wx, m, 32);
            wy += __shfl_xor(wy, m, 32);
        }

        // inclusive prefix of row sums over the 16 rows (within each lane half)
        float px = rx, py = ry;
        #pragma unroll
        for (int d = 1; d < 16; d <<= 1) {
            float tx = __shfl_up(px, d, 16);
            float ty = __shfl_up(py, d, 16);
            if (lane16 >= d) { px += tx; py += ty; }
        }

        // ---- exchange wave totals across the 8 waves (double-buffered LDS) ----
        const int bufi = it & 1;
        if (lane == 0) { wsx[bufi][wave] = wx; wsy[bufi][wave] = wy; }
        __syncthreads();
        float woffx = carry_x, woffy = carry_y;   // offset of this wave's tile
        float totx  = carry_x, toty  = carry_y;   // next carry
        #pragma unroll
        for (int w = 0; w < 8; ++w) {
            float sxv = wsx[bufi][w], syv = wsy[bufi][w];
            if (w < wave) { woffx += sxv; woffy += syv; }
            totx += sxv; toty += syv;
        }
        carry_x = totx; carry_y = toty;

        // ---- seed C: vel[0] + carry + wave offset + exclusive row offset ----
        v8f dx, dy;
        #pragma unroll
        for (int rr = 0; rr < 8; ++rr) {
            int row = rowbase + rr;
            int src = (row > 0) ? (row - 1) : 0;
            float ox = __shfl(px, src, 16);
            float oy = __shfl(py, src, 16);
            if (row == 0) { ox = 0.f; oy = 0.f; }
            dx[rr] = v0.x + woffx + ox;
            dy[rr] = v0.y + woffy + oy;
        }

        // ---- D = M * U' + C : 4 chained V_WMMA_F32_16X16X4_F32 per component ----
        #pragma unroll
        for (int c = 0; c < 4; ++c) {
            const int k0 = 4 * c + khalf;
            v2f bm;                                     // strict upper-tri ones: B[k][n]=(k<n)
            bm.x = (k0     < lane16) ? 1.f : 0.f;
            bm.y = (k0 + 1 < lane16) ? 1.f : 0.f;
            dx = __builtin_amdgcn_wmma_f32_16x16x4_f32(false, ax[c], false, bm,
                                                       (short)0, dx, false, false);
            dy = __builtin_amdgcn_wmma_f32_16x16x4_f32(false, ay[c], false, bm,
                                                       (short)0, dy, false, false);
        }

        // ---- position loss from D (C/D layout): pos_pred = DT * D ----
        #pragma unroll
        for (int rr = 0; rr < 8; ++rr) {
            int row = rowbase + rr;
            int t = tb + 16 * row + lane16;
            v2f pg = __builtin_nontemporal_load(pg2 + base + (size_t)t);
            float ex = MTL_DT * dx[rr] - pg.x;
            float ey = MTL_DT * dy[rr] - pg.y;
            accp += ex * ex + ey * ey;
        }
    }

    // ---- fixed-order block reduction -> 4 partials per block ----
    __shared__ v4f red[256];
    v4f mine; mine.x = accp; mine.y = acce; mine.z = accml; mine.w = accreg;
    red[tid] = mine;
    __syncthreads();
    #pragma unroll
    for (int s = 128; s > 0; s >>= 1) {
        if (tid < s) {
            v4f o = red[tid + s];
            v4f m = red[tid];
            m.x += o.x; m.y += o.y; m.z += o.z; m.w += o.w;
            red[tid] = m;
        }
        __syncthreads();
    }
    if (tid == 0) __builtin_nontemporal_store(red[0], (v4f*)ws + b);
}

__global__ __launch_bounds__(256) void mtl_final_kernel(
    const float* __restrict__ ws,
    const float* __restrict__ log_delta_v,
    const float* __restrict__ log_delta_c,
    float* __restrict__ out)
{
    __shared__ v4f red[256];
    const int tid = threadIdx.x;
    red[tid] = ((const v4f*)ws)[tid];
    __syncthreads();
    #pragma unroll
    for (int s = 128; s > 0; s >>= 1) {
        if (tid < s) {
            v4f o = red[tid + s];
            v4f m = red[tid];
            m.x += o.x; m.y += o.y; m.z += o.z; m.w += o.w;
            red[tid] = m;
        }
        __syncthreads();
    }
    if (tid == 0) {
        v4f s = red[0];
        const float Np  = 8388608.f;   // B*T*2
        const float Nbt = 4194304.f;   // B*T
        float lv  = (s.x + s.y) / Np;
        float lc  = 0.5f * (s.z / Nbt);
        float ldv = log_delta_v[0], ldc = log_delta_c[0];
        float dv  = __expf(ldv), dc = __expf(ldc);
        float total = lv / (2.f * dv * dv) + lc / (2.f * dc * dc)
                    + ldv + ldc + 0.01f * (s.w / Nbt);
        out[0] = total;
    }
}

extern "C" void kernel_launch(void* const* d_in, const int* in_sizes, int n_in,
                              void* d_out, int out_size, void* d_ws, size_t ws_size,
                              hipStream_t stream) {
    const float* vel_pred = (const float*)d_in[0];
    const float* cov_pred = (const float*)d_in[1];
    const float* vel_gt   = (const float*)d_in[2];
    const float* pos_gt   = (const float*)d_in[3];
    const float* ldv      = (const float*)d_in[4];
    const float* ldc      = (const float*)d_in[5];
    float* ws = (float*)d_ws;   // 256 blocks * 4 floats of partials = 4 KiB

    mtl_main_kernel<<<B_DIM, 256, 0, stream>>>(vel_pred, cov_pred, vel_gt, pos_gt, ws);
    mtl_final_kernel<<<1, 256, 0, stream>>>(ws, ldv, ldc, (float*)d_out);
}